// Seq2SeqModel_44109314130180
// MI455X (gfx1250) — compile-verified
//
#include <hip/hip_runtime.h>
#include <math.h>

#define H    512
#define B    64
#define TSRC 512
#define TTGT 256
#define NEG_INF -1e9f

typedef __attribute__((ext_vector_type(2))) float v2f;
typedef __attribute__((ext_vector_type(8))) float v8f;

// ---------------------------------------------------------------------------
// Multi-term fp32 WMMA GEMM:  Z[M,N] = sum_t A_t[M,K_t](lda_t) @ W_t[K_t,N] + bias
// Uses V_WMMA_F32_16X16X4_F32 (wave32, 16x16 C tile per wave, 4 N-subtiles/wave
// for A-fragment reuse). Optional second store target (ldz2) for d_out slice.
//
// Fragment layouts per CDNA5 ISA 7.12.2:
//  A 16x4 f32: lane l -> m=l&15, half=l>>4; a.x=A[m,k+2*half], a.y=A[m,k+2*half+1]
//  B 4x16 f32: lane l -> n=l&15, half=l>>4; b.x=W[k+2*half,n], b.y=W[k+2*half+1,n]
//  C 16x16 f32: acc[v] = C[v + 8*half, n]
// ---------------------------------------------------------------------------
__global__ __launch_bounds__(256)
void gemm_f32_wmma(const float* __restrict__ A0, int lda0, const float* __restrict__ W0, int K0,
                   const float* __restrict__ A1, int lda1, const float* __restrict__ W1, int K1,
                   const float* __restrict__ A2, int lda2, const float* __restrict__ W2, int K2,
                   const float* __restrict__ bias,
                   float* __restrict__ Z,
                   float* __restrict__ Z2, long long ldz2,
                   int M, int N)
{
    const int wave   = (int)((blockIdx.x * blockDim.x + threadIdx.x) >> 5);
    const int lane   = threadIdx.x & 31;
    const int tilesN = N >> 6;                  // 64-wide N supertiles
    if (wave >= (M >> 4) * tilesN) return;
    const int tM   = wave / tilesN;
    const int tN   = wave % tilesN;
    const int m0   = tM << 4;
    const int n0   = tN << 6;
    const int half = lane >> 4;
    const int l    = lane & 15;

    v8f acc0 = {}, acc1 = {}, acc2 = {}, acc3 = {};

    const float* Aarr[3] = {A0, A1, A2};
    const float* Warr[3] = {W0, W1, W2};
    const int    Karr[3] = {K0, K1, K2};
    const int    Larr[3] = {lda0, lda1, lda2};

    for (int t = 0; t < 3; ++t) {
        const float* A = Aarr[t];
        if (!A) continue;
        const float* W   = Warr[t];
        const int    K   = Karr[t];
        const int    lda = Larr[t];
        const float* arow = A + (long long)(m0 + l) * lda + 2 * half;

        #pragma unroll 4
        for (int k = 0; k < K; k += 4) {
            v2f a = *(const v2f*)(arow + k);
            const float* w0r = W + (long long)(k + 2 * half) * N + n0 + l;
            const float* w1r = w0r + N;

            v2f b;
            b.x = w0r[0];  b.y = w1r[0];
            acc0 = __builtin_amdgcn_wmma_f32_16x16x4_f32(false, a, false, b, (short)0, acc0, false, false);
            b.x = w0r[16]; b.y = w1r[16];
            acc1 = __builtin_amdgcn_wmma_f32_16x16x4_f32(false, a, false, b, (short)0, acc1, false, false);
            b.x = w0r[32]; b.y = w1r[32];
            acc2 = __builtin_amdgcn_wmma_f32_16x16x4_f32(false, a, false, b, (short)0, acc2, false, false);
            b.x = w0r[48]; b.y = w1r[48];
            acc3 = __builtin_amdgcn_wmma_f32_16x16x4_f32(false, a, false, b, (short)0, acc3, false, false);
        }
    }

    v8f accs[4] = {acc0, acc1, acc2, acc3};
    #pragma unroll
    for (int j = 0; j < 4; ++j) {
        const int n  = n0 + 16 * j + l;
        const float bv = bias ? bias[n] : 0.f;
        #pragma unroll
        for (int v = 0; v < 8; ++v) {
            const int m = m0 + v + 8 * half;
            const float r = accs[j][v] + bv;
            Z[(long long)m * N + n] = r;
            if (Z2) Z2[(long long)m * ldz2 + n] = r;
        }
    }
}

// ---------------------------------------------------------------------------
// LSTM gate nonlinearity: z[B,4H] (i,f,j,o order), updates h,c in place.
// ---------------------------------------------------------------------------
__global__ __launch_bounds__(256)
void lstm_gates(const float* __restrict__ z, float* __restrict__ h, float* __restrict__ c)
{
    const int idx = blockIdx.x * blockDim.x + threadIdx.x;
    if (idx >= B * H) return;
    const int b = idx >> 9;          // /H
    const int n = idx & (H - 1);
    const float* zb = z + (long long)b * 4 * H;
    const float zi = zb[n];
    const float zf = zb[n + H];
    const float zj = zb[n + 2 * H];
    const float zo = zb[n + 3 * H];
    const float si = 1.f / (1.f + __expf(-zi));
    const float sf = 1.f / (1.f + __expf(-zf));
    const float so = 1.f / (1.f + __expf(-zo));
    const float cn = sf * c[idx] + si * tanhf(zj);
    c[idx] = cn;
    h[idx] = so * tanhf(cn);
}

// ---------------------------------------------------------------------------
// Init recurrent state from inputs; zero attn feedback.
// ---------------------------------------------------------------------------
__global__ __launch_bounds__(256)
void init_state(const float* __restrict__ fwh, const float* __restrict__ fwc,
                const float* __restrict__ bwh, const float* __restrict__ bwc,
                float* __restrict__ h1, float* __restrict__ c1,
                float* __restrict__ h2, float* __restrict__ c2,
                float* __restrict__ attn)
{
    const int i = blockIdx.x * blockDim.x + threadIdx.x;
    if (i >= B * H) return;
    h1[i] = fwh[i]; c1[i] = fwc[i];
    h2[i] = bwh[i]; c2[i] = bwc[i];
    attn[i] = 0.f;
}

// ---------------------------------------------------------------------------
// qm[b,k] = sum_h W_memory[k,h] * h2[b,h]   (the score trick: enc·(Wm@h2))
// ---------------------------------------------------------------------------
__global__ __launch_bounds__(256)
void memory_query(const float* __restrict__ Wm,   // [2H,H]
                  const float* __restrict__ h2,   // [B,H]
                  float* __restrict__ qm)         // [B,2H]
{
    const int idx = blockIdx.x * blockDim.x + threadIdx.x;
    if (idx >= B * 2 * H) return;
    const int b = idx >> 10;            // /(2H)
    const int k = idx & (2 * H - 1);
    const float4* w = (const float4*)(Wm + (long long)k * H);
    const float4* q = (const float4*)(h2 + (long long)b * H);
    float acc = 0.f;
    #pragma unroll 4
    for (int i = 0; i < H / 4; ++i) {
        float4 wv = w[i], qv = q[i];
        acc += wv.x * qv.x + wv.y * qv.y + wv.z * qv.z + wv.w * qv.w;
    }
    qm[idx] = acc;
}

// ---------------------------------------------------------------------------
// Fused scores + softmax + context, one block per batch row.
// scores[s] = enc[b,s,:]·qm[b,:] + mask[b,s]*NEG_INF ; ctx = softmax(scores)@enc[b]
// encoder_outputs (134MB) fits MI455X's 192MB L2 -> both passes stream from L2.
// ---------------------------------------------------------------------------
__global__ __launch_bounds__(256)
void attention_ctx(const float* __restrict__ enc,   // [B,S,2H]
                   const float* __restrict__ mask,  // [B,S]
                   const float* __restrict__ qm,    // [B,2H]
                   float* __restrict__ ctx)         // [B,2H]
{
    __shared__ float sh_q[2 * H];
    __shared__ float sh_s[TSRC];
    __shared__ float red[256];

    const int b = blockIdx.x, tid = threadIdx.x;
    for (int i = tid; i < 2 * H; i += 256) sh_q[i] = qm[(long long)b * 2 * H + i];
    __syncthreads();

    const float* encb = enc + (long long)b * TSRC * 2 * H;

    for (int s = tid; s < TSRC; s += 256) {
        const float4* row = (const float4*)(encb + (long long)s * 2 * H);
        float acc = 0.f;
        #pragma unroll 4
        for (int k = 0; k < (2 * H) / 4; ++k) {
            float4 e = row[k];
            acc += e.x * sh_q[4 * k] + e.y * sh_q[4 * k + 1]
                 + e.z * sh_q[4 * k + 2] + e.w * sh_q[4 * k + 3];
        }
        sh_s[s] = acc + mask[(long long)b * TSRC + s] * NEG_INF;
    }
    __syncthreads();

    float mx = -INFINITY;
    for (int s = tid; s < TSRC; s += 256) mx = fmaxf(mx, sh_s[s]);
    red[tid] = mx; __syncthreads();
    for (int off = 128; off > 0; off >>= 1) {
        if (tid < off) red[tid] = fmaxf(red[tid], red[tid + off]);
        __syncthreads();
    }
    mx = red[0]; __syncthreads();

    float sum = 0.f;
    for (int s = tid; s < TSRC; s += 256) {
        float e = __expf(sh_s[s] - mx);
        sh_s[s] = e;
        sum += e;
    }
    red[tid] = sum; __syncthreads();
    for (int off = 128; off > 0; off >>= 1) {
        if (tid < off) red[tid] += red[tid + off];
        __syncthreads();
    }
    const float inv = 1.f / red[0]; __syncthreads();
    for (int s = tid; s < TSRC; s += 256) sh_s[s] *= inv;
    __syncthreads();

    for (int k = tid; k < 2 * H; k += 256) {
        float acc = 0.f;
        #pragma unroll 4
        for (int s = 0; s < TSRC; ++s)
            acc += sh_s[s] * encb[(long long)s * 2 * H + k];
        ctx[(long long)b * 2 * H + k] = acc;
    }
}

// ---------------------------------------------------------------------------
extern "C" void kernel_launch(void* const* d_in, const int* in_sizes, int n_in,
                              void* d_out, int out_size, void* d_ws, size_t ws_size,
                              hipStream_t stream)
{
    const float* tgt  = (const float*)d_in[0];   // [B,TTGT,H]
    const float* fwh  = (const float*)d_in[1];
    const float* fwc  = (const float*)d_in[2];
    const float* bwh  = (const float*)d_in[3];
    const float* bwc  = (const float*)d_in[4];
    const float* enc  = (const float*)d_in[5];   // [B,TSRC,2H]
    const float* mask = (const float*)d_in[6];   // [B,TSRC]
    const float* Wm   = (const float*)d_in[7];   // [2H,H]
    const float* Wa   = (const float*)d_in[8];   // [3H,H]
    const float* ki1  = (const float*)d_in[9];   // [2H,4H]
    const float* kr1  = (const float*)d_in[10];  // [H,4H]
    const float* b1   = (const float*)d_in[11];  // [4H]
    const float* ki2  = (const float*)d_in[12];  // [H,4H]
    const float* kr2  = (const float*)d_in[13];  // [H,4H]
    const float* b2   = (const float*)d_in[14];  // [4H]
    float* out = (float*)d_out;                  // [B,TTGT,H]

    float* ws   = (float*)d_ws;
    float* z    = ws;                 // B*4H
    float* h1   = z    + B * 4 * H;   // B*H
    float* c1   = h1   + B * H;
    float* h2   = c1   + B * H;
    float* c2   = h2   + B * H;
    float* attn = c2   + B * H;
    float* ctx  = attn + B * H;       // B*2H
    float* qm   = ctx  + B * 2 * H;   // B*2H

    init_state<<<(B * H + 255) / 256, 256, 0, stream>>>(fwh, fwc, bwh, bwc, h1, c1, h2, c2, attn);

    const int blkGemmZ = ((B / 16) * ((4 * H) / 64) + 7) / 8;   // 128 waves -> 16 blocks
    const int blkGemmA = ((B / 16) * (H / 64) + 7) / 8;         // 32 waves  -> 4 blocks

    for (int t = 0; t < TTGT; ++t) {
        const float* x_t = tgt + (long long)t * H;              // row b: + b*TTGT*H

        // z1 = x_t@ki1[0:H] + attn@ki1[H:2H] + h1@kr1 + b1
        gemm_f32_wmma<<<blkGemmZ, 256, 0, stream>>>(
            x_t,  TTGT * H, ki1,                        H,
            attn, H,        ki1 + (long long)H * 4 * H, H,
            h1,   H,        kr1,                        H,
            b1, z, nullptr, 0, B, 4 * H);
        lstm_gates<<<(B * H + 255) / 256, 256, 0, stream>>>(z, h1, c1);

        // z2 = h1@ki2 + h2@kr2 + b2
        gemm_f32_wmma<<<blkGemmZ, 256, 0, stream>>>(
            h1, H, ki2, H,
            h2, H, kr2, H,
            nullptr, 0, nullptr, 0,
            b2, z, nullptr, 0, B, 4 * H);
        lstm_gates<<<(B * H + 255) / 256, 256, 0, stream>>>(z, h2, c2);

        // attention (score trick + fused softmax/context)
        memory_query<<<(B * 2 * H + 255) / 256, 256, 0, stream>>>(Wm, h2, qm);
        attention_ctx<<<B, 256, 0, stream>>>(enc, mask, qm, ctx);

        // attn = [h2, ctx] @ Wa  -> recurrent buffer and d_out[:, t, :]
        gemm_f32_wmma<<<blkGemmA, 256, 0, stream>>>(
            h2,  H,     Wa,                    H,
            ctx, 2 * H, Wa + (long long)H * H, 2 * H,
            nullptr, 0, nullptr, 0,
            nullptr, attn, out + (long long)t * H, (long long)TTGT * H, B, H);
    }
}